// Conv1dFFTInt8_88622355186007
// MI455X (gfx1250) — compile-verified
//
#include <hip/hip_runtime.h>
#include <stdint.h>

// Problem constants (from reference)
#define B_    16
#define CIN_  128
#define COUT_ 128
#define L_    4096
#define KTOT  (CIN_ * L_)      // 524288 reduction length
#define KC    1024             // K chunk per workgroup (split-K)
#define NSPLIT (KTOT / KC)     // 512 workgroups
#define KT    64               // inner LDS tile along K
#define NTILES (KC / KT)       // 16 inner tiles (even -> clean double buffering)
#define LDA   68               // padded LDS row stride (floats): 272B, 16B aligned, bank-conflict free
#define LDW   68
#define AOFF  0                // A tile offset (floats) within one buffer
#define WOFF  (16 * LDA)       // W tile offset (floats) within one buffer
#define LDSF  (16 * LDA + COUT_ * LDW)   // 9792 floats per buffer

typedef __attribute__((ext_vector_type(2))) float v2f;
typedef __attribute__((ext_vector_type(8))) float v8f;

__global__ __launch_bounds__(256) void conv1dfft_bias_init(const float* __restrict__ bias,
                                                           float* __restrict__ out) {
  int idx = blockIdx.x * 256 + threadIdx.x;
  if (idx < B_ * COUT_) out[idx] = bias[idx & (COUT_ - 1)];
}

// out[b, o] += sum over this block's K-chunk of  w[o, k] * x_rev[b, k]
// where k = ci*L + l and x_rev picks x[b, ci, (L - l) & (L-1)].
__global__ __launch_bounds__(256) void conv1dfft_gemm(const float* __restrict__ x,
                                                      const float* __restrict__ w,
                                                      float* __restrict__ out) {
  __shared__ float lds[2 * LDSF];        // double-buffered tiles, ~76.5 KB

  const int tid  = threadIdx.x;
  const int lane = tid & 31;
  const int wv   = tid >> 5;             // wave id 0..7 -> COUT slice n0 = wv*16
  const int s    = blockIdx.x;           // K-chunk id, 0..NSPLIT-1
  const int ci   = s >> 2;               // KC=1024 -> 4 chunks per cin row
  const int l0   = (s & 3) * KC;

  const float* xb = x + (size_t)ci * L_;       // + b*KTOT + reversed col
  const float* wb = w + (size_t)ci * L_ + l0;  // + o*KTOT + local k

  // Issue asynchronous global->LDS copies for tile `kt` into buffer `buf`.
  // A tile: 16x64 via per-lane b32 (handles the index reversal / wrap);
  // W tile: 128x64 via per-lane b128 (coalesced 256B rows).
  auto issue_tile = [&](int kt, int buf) {
    float* base = &lds[buf * LDSF];
#pragma unroll
    for (int j = 0; j < 4; ++j) {
      int e   = tid + j * 256;
      int row = e >> 6;                         // batch b
      int col = e & 63;
      int l   = l0 + kt + col;
      int src = (L_ - l) & (L_ - 1);            // reversal from the FFT identity
      const float* g = xb + (size_t)row * KTOT + src;
      unsigned lo = (unsigned)(uintptr_t)(base + AOFF + row * LDA + col);
      asm volatile("global_load_async_to_lds_b32 %0, %1, off"
                   :: "v"(lo), "v"(g) : "memory");
    }
#pragma unroll
    for (int j = 0; j < 8; ++j) {
      int ve   = tid + j * 256;
      int row  = ve >> 4;                       // cout o
      int col4 = ve & 15;
      const float* g = wb + (size_t)row * KTOT + kt + col4 * 4;
      unsigned lo = (unsigned)(uintptr_t)(base + WOFF + row * LDW + col4 * 4);
      asm volatile("global_load_async_to_lds_b128 %0, %1, off"
                   :: "v"(lo), "v"(g) : "memory");
    }
  };

  v8f acc = {0.f, 0.f, 0.f, 0.f, 0.f, 0.f, 0.f, 0.f};

  // WMMA operand addressing (ISA 7.12.2, 32-bit 16x4 A / 4x16 B layouts):
  //   A: lane l, vgpr v -> (m = l&15,       k = (l>>4)*2 + v)
  //   B: lane l, vgpr v -> (k = (l>>4)*2+v, n = l&15)
  const int m  = lane & 15;
  const int kh = (lane >> 4) << 1;

  issue_tile(0, 0);
  for (int t = 0; t < NTILES; ++t) {
    asm volatile("s_wait_asynccnt 0x0" ::: "memory");  // my tile-t copies done
    __syncthreads();                                   // everyone's tile-t copies done
    if (t + 1 < NTILES) issue_tile((t + 1) * KT, (t + 1) & 1);

    const float* buf = &lds[(t & 1) * LDSF];
    const float* ap  = buf + AOFF + m * LDA + kh;
    const float* bp  = buf + WOFF + (wv * 16 + m) * LDW + kh;
#pragma unroll
    for (int kk = 0; kk < KT; kk += 4) {
      v2f a = *(const v2f*)(ap + kk);
      v2f b = *(const v2f*)(bp + kk);
      acc = __builtin_amdgcn_wmma_f32_16x16x4_f32(
          /*neg_a=*/false, a, /*neg_b=*/false, b,
          /*c_mod=*/(short)0, acc, /*reuse_a=*/false, /*reuse_b=*/false);
    }
    // No trailing barrier needed: the barrier at the top of iteration t+1
    // orders all waves' reads of this buffer before it is re-issued at t+2.
  }

  // C/D layout: lane l, vgpr v -> (m = v + (l>>4)*8, n = l&15). Split-K partial -> atomic add.
  const int n     = wv * 16 + (lane & 15);
  const int mbase = (lane >> 4) * 8;
#pragma unroll
  for (int v = 0; v < 8; ++v) {
    atomicAdd(&out[(mbase + v) * COUT_ + n], acc[v]);
  }
}

extern "C" void kernel_launch(void* const* d_in, const int* in_sizes, int n_in,
                              void* d_out, int out_size, void* d_ws, size_t ws_size,
                              hipStream_t stream) {
  const float* x    = (const float*)d_in[0];   // [16, 128, 4096]
  const float* wgt  = (const float*)d_in[1];   // [128, 128, 4096]
  const float* bias = (const float*)d_in[2];   // [128]
  float* out = (float*)d_out;                  // [16, 128, 1] -> 2048 floats

  conv1dfft_bias_init<<<(B_ * COUT_ + 255) / 256, 256, 0, stream>>>(bias, out);
  conv1dfft_gemm<<<NSPLIT, 256, 0, stream>>>(x, wgt, out);
}